// GraphSAGE_11381663334735
// MI455X (gfx1250) — compile-verified
//
#include <hip/hip_runtime.h>
#include <math.h>

#define Nn 100000
#define Ne 1600000
// IN_C=64, HID_C=128, OUT_C=64

typedef __attribute__((ext_vector_type(2))) float v2f;
typedef __attribute__((ext_vector_type(8))) float v8f;

__device__ __forceinline__ v8f wmma4(v2f a, v2f b, v8f c) {
  // D = A(16x4 f32) * B(4x16 f32) + C(16x16 f32)
  return __builtin_amdgcn_wmma_f32_16x16x4_f32(false, a, false, b, 0, c, false, false);
}

// Accumulate A(16xK, in LDS with row stride KP) @ W(KxNCOLS, global row-major)
// into a 16x16 WMMA accumulator at column offset colbase.
template <int K, int KP, int NCOLS>
__device__ __forceinline__ void gemm_tile(const float* Alds, const float* __restrict__ W,
                                          int colbase, int lane, v8f& acc) {
  const int m  = lane & 15;          // A row
  const int k2 = (lane >> 4) * 2;    // K sub-offset for this lane half
  const int n  = lane & 15;          // B col
#pragma unroll
  for (int k = 0; k < K; k += 4) {
    v2f a, b;
    a.x = Alds[m * KP + k + k2];
    a.y = Alds[m * KP + k + k2 + 1];
    b.x = W[(size_t)(k + k2) * NCOLS + colbase + n];
    b.y = W[(size_t)(k + k2 + 1) * NCOLS + colbase + n];
    acc = wmma4(a, b, acc);
  }
}

// ---------------- degree / scatter kernels ----------------

__global__ void k_degree(const int* __restrict__ dst, float* __restrict__ deg) {
  int e = blockIdx.x * blockDim.x + threadIdx.x;
  if (e < Ne) atomicAdd(&deg[dst[e]], 1.0f);
}

__global__ void k_invdeg(float* __restrict__ deg) {
  int i = blockIdx.x * blockDim.x + threadIdx.x;
  if (i < Nn) {
    float d = deg[i];
    deg[i] = 1.0f / fmaxf(d, 1.0f);
  }
}

// one wave per edge; lane covers 2 channels of the 64-wide row
__global__ void k_scatter64(const int* __restrict__ src, const int* __restrict__ dst,
                            const float* __restrict__ x, float* __restrict__ agg) {
  unsigned gid  = blockIdx.x * blockDim.x + threadIdx.x;
  unsigned e    = gid >> 5;
  unsigned lane = gid & 31;
  if (e >= Ne) return;
  int s = src[e], d = dst[e];
  float2 v = *(const float2*)(x + (size_t)s * 64 + lane * 2);
  float* a = agg + (size_t)d * 64 + lane * 2;
  atomicAdd(a + 0, v.x);
  atomicAdd(a + 1, v.y);
}

// one wave per edge; lane covers 4 channels of the 128-wide row
__global__ void k_scatter128(const int* __restrict__ src, const int* __restrict__ dst,
                             const float* __restrict__ h, float* __restrict__ agg) {
  unsigned gid  = blockIdx.x * blockDim.x + threadIdx.x;
  unsigned e    = gid >> 5;
  unsigned lane = gid & 31;
  if (e >= Ne) return;
  int s = src[e], d = dst[e];
  float4 v = *(const float4*)(h + (size_t)s * 128 + lane * 4);
  float* a = agg + (size_t)d * 128 + lane * 4;
  atomicAdd(a + 0, v.x);
  atomicAdd(a + 1, v.y);
  atomicAdd(a + 2, v.z);
  atomicAdd(a + 3, v.w);
}

// ---------------- conv1: h = relu(agg/deg @ Wl1 + bl1 + x @ Wr1) ----------------

__global__ void __launch_bounds__(128) k_sage1(
    const float* __restrict__ agg, const float* __restrict__ invdeg,
    const float* __restrict__ x,
    const float* __restrict__ Wl, const float* __restrict__ bl,
    const float* __restrict__ Wr, float* __restrict__ h) {
  __shared__ __align__(16) float Aag[16 * 68];  // padded stride 68
  __shared__ __align__(16) float Ax [16 * 68];
  const int rowbase = blockIdx.x * 16;
  const int t = threadIdx.x;
  // stage 16x64 x2 (256 float4 each)
  for (int i = t; i < 256; i += 128) {
    int r = i >> 4, c4 = i & 15;
    int row = rowbase + r;
    float sc = invdeg[row];
    float4 va = *(const float4*)(agg + (size_t)row * 64 + c4 * 4);
    float4 vx = *(const float4*)(x   + (size_t)row * 64 + c4 * 4);
    float4 vs; vs.x = va.x * sc; vs.y = va.y * sc; vs.z = va.z * sc; vs.w = va.w * sc;
    *(float4*)(Aag + r * 68 + c4 * 4) = vs;
    *(float4*)(Ax  + r * 68 + c4 * 4) = vx;
  }
  __syncthreads();
  const int lane = t & 31, wave = t >> 5;
  const int n = lane & 15, kg = lane >> 4;
#pragma unroll
  for (int ct = 0; ct < 2; ++ct) {
    int colbase = (wave + ct * 4) * 16;  // 8 col-tiles over 128 cols, 2 per wave
    v8f acc = {};
    gemm_tile<64, 68, 128>(Aag, Wl, colbase, lane, acc);
    gemm_tile<64, 68, 128>(Ax,  Wr, colbase, lane, acc);
    float b = bl[colbase + n];
#pragma unroll
    for (int i = 0; i < 8; ++i) {
      float v = acc[i] + b;
      v = v > 0.0f ? v : 0.0f;
      h[(size_t)(rowbase + i + 8 * kg) * 128 + colbase + n] = v;
    }
  }
}

// -------- conv2 + MLP head fused:
// emb = agg/deg @ Wl2 + bl2 + h @ Wr2
// c1 = relu(emb @ Wc1 + bc1); c2 = relu(c1 @ Wc2 + bc2); p = sigmoid(c2 @ Wc3 + bc3)

__global__ void __launch_bounds__(128) k_sage2_mlp(
    const float* __restrict__ agg, const float* __restrict__ invdeg,
    const float* __restrict__ h,
    const float* __restrict__ Wl, const float* __restrict__ bl,
    const float* __restrict__ Wr,
    const float* __restrict__ Wc1, const float* __restrict__ bc1,
    const float* __restrict__ Wc2, const float* __restrict__ bc2,
    const float* __restrict__ Wc3, const float* __restrict__ bc3,
    float* __restrict__ emb_out, float* __restrict__ prob_out) {
  __shared__ __align__(16) float A0[16 * 132];  // scaled agg rows
  __shared__ __align__(16) float A1[16 * 132];  // h rows
  __shared__ __align__(16) float EE[16 * 68];   // embeddings tile
  __shared__ __align__(16) float C1[16 * 132];
  __shared__ __align__(16) float C2[16 * 68];
  const int rowbase = blockIdx.x * 16;
  const int t = threadIdx.x, lane = t & 31, wave = t >> 5;
  const int n = lane & 15, kg = lane >> 4;
  // stage 16x128 x2 (512 float4 each)
  for (int i = t; i < 512; i += 128) {
    int r = i >> 5, c4 = i & 31;
    int row = rowbase + r;
    float sc = invdeg[row];
    float4 va = *(const float4*)(agg + (size_t)row * 128 + c4 * 4);
    float4 vh = *(const float4*)(h   + (size_t)row * 128 + c4 * 4);
    float4 vs; vs.x = va.x * sc; vs.y = va.y * sc; vs.z = va.z * sc; vs.w = va.w * sc;
    *(float4*)(A0 + r * 132 + c4 * 4) = vs;
    *(float4*)(A1 + r * 132 + c4 * 4) = vh;
  }
  __syncthreads();
  // emb tile: 16x64, one col-tile per wave, K=128
  {
    int colbase = wave * 16;
    v8f acc = {};
    gemm_tile<128, 132, 64>(A0, Wl, colbase, lane, acc);
    gemm_tile<128, 132, 64>(A1, Wr, colbase, lane, acc);
    float b = bl[colbase + n];
#pragma unroll
    for (int i = 0; i < 8; ++i) {
      float v = acc[i] + b;
      int r = i + 8 * kg;
      EE[r * 68 + colbase + n] = v;
      emb_out[(size_t)(rowbase + r) * 64 + colbase + n] = v;
    }
  }
  __syncthreads();
  // c1 = relu(emb @ Wc1 + bc1): 16x128, 2 tiles per wave, K=64
#pragma unroll
  for (int ct = 0; ct < 2; ++ct) {
    int colbase = (wave + ct * 4) * 16;
    v8f acc = {};
    gemm_tile<64, 68, 128>(EE, Wc1, colbase, lane, acc);
    float b = bc1[colbase + n];
#pragma unroll
    for (int i = 0; i < 8; ++i) {
      float v = acc[i] + b;
      C1[(i + 8 * kg) * 132 + colbase + n] = v > 0.0f ? v : 0.0f;
    }
  }
  __syncthreads();
  // c2 = relu(c1 @ Wc2 + bc2): 16x64, one tile per wave, K=128
  {
    int colbase = wave * 16;
    v8f acc = {};
    gemm_tile<128, 132, 64>(C1, Wc2, colbase, lane, acc);
    float b = bc2[colbase + n];
#pragma unroll
    for (int i = 0; i < 8; ++i) {
      float v = acc[i] + b;
      C2[(i + 8 * kg) * 68 + colbase + n] = v > 0.0f ? v : 0.0f;
    }
  }
  __syncthreads();
  // probs = sigmoid(c2 @ Wc3 + bc3): 16 rows, 64-dot each
  if (t < 16) {
    float s = bc3[0];
#pragma unroll 8
    for (int j = 0; j < 64; ++j) s += C2[t * 68 + j] * Wc3[j];
    prob_out[rowbase + t] = 1.0f / (1.0f + expf(-s));
  }
}

// ---------------- launch ----------------

extern "C" void kernel_launch(void* const* d_in, const int* in_sizes, int n_in,
                              void* d_out, int out_size, void* d_ws, size_t ws_size,
                              hipStream_t stream) {
  const float* x   = (const float*)d_in[0];
  const int*   ei  = (const int*)d_in[1];   // [2, E] -> src rows then dst rows
  const float* Wl1 = (const float*)d_in[2];
  const float* bl1 = (const float*)d_in[3];
  const float* Wr1 = (const float*)d_in[4];
  const float* Wl2 = (const float*)d_in[5];
  const float* bl2 = (const float*)d_in[6];
  const float* Wr2 = (const float*)d_in[7];
  const float* Wc1 = (const float*)d_in[8];
  const float* bc1 = (const float*)d_in[9];
  const float* Wc2 = (const float*)d_in[10];
  const float* bc2 = (const float*)d_in[11];
  const float* Wc3 = (const float*)d_in[12];
  const float* bc3 = (const float*)d_in[13];
  const int* src = ei;
  const int* dst = ei + Ne;

  // workspace layout (bytes): deg[N] | agg (N*128, reused for both layers) | h (N*128)
  char* ws = (char*)d_ws;
  float* deg = (float*)(ws);
  float* agg = (float*)(ws + 401408);                          // 16B aligned
  float* hbuf = (float*)(ws + 401408 + (size_t)Nn * 128 * 4);
  float* emb  = (float*)d_out;
  float* prob = (float*)d_out + (size_t)Nn * 64;

  hipMemsetAsync(deg, 0, (size_t)Nn * 4, stream);
  hipMemsetAsync(agg, 0, (size_t)Nn * 64 * 4, stream);

  k_degree<<<(Ne + 255) / 256, 256, 0, stream>>>(dst, deg);
  k_invdeg<<<(Nn + 255) / 256, 256, 0, stream>>>(deg);

  k_scatter64<<<Ne / 8, 256, 0, stream>>>(src, dst, x, agg);   // E waves
  k_sage1<<<Nn / 16, 128, 0, stream>>>(agg, deg, x, Wl1, bl1, Wr1, hbuf);

  hipMemsetAsync(agg, 0, (size_t)Nn * 128 * 4, stream);        // reuse as agg2
  k_scatter128<<<Ne / 8, 256, 0, stream>>>(src, dst, hbuf, agg);
  k_sage2_mlp<<<Nn / 16, 128, 0, stream>>>(agg, deg, hbuf, Wl2, bl2, Wr2,
                                           Wc1, bc1, Wc2, bc2, Wc3, bc3, emb, prob);
}